// BilateralFilter_69080253989525
// MI455X (gfx1250) — compile-verified
//
#include <hip/hip_runtime.h>
#include <cstdint>
#include <cstddef>

// Bilateral 7x7, sigma_space=5, sigma_color=0.1, zero padding.
// Image: (N, 1, 720, 1280) fp32 -> out (N, 720, 1280) fp32.

#define IMG_W 1280
#define IMG_H 720
#define TX 64              // output tile width
#define TY 16              // output tile height
#define HALO 3
#define XMARG 16           // left margin (floats) so every 4-float chunk is 16B-aligned
#define LW 84              // tile width incl. margins: 16 + 64 + 4 (= 21 x 4-float chunks)
#define NCHUNK (LW / 4)    // 21 b128 chunks per row
#define LH (TY + 2 * HALO) // 22 rows incl. halo
#define LSTRIDE 84         // LDS row stride in floats (336 B, 16B aligned, 20-bank skew)
#define NTHREADS 256
#define TOTAL_CHUNKS (LH * NCHUNK)  // 462

#if defined(__HIP_DEVICE_COMPILE__) && __has_builtin(__builtin_amdgcn_exp2f)
#define FAST_EXP2(x) __builtin_amdgcn_exp2f(x)
#else
#define FAST_EXP2(x) exp2f(x)
#endif

// exp(-d2/(2*0.1^2)) * exp(-r2/(2*5^2)) = exp2(d2*K_COLOR + r2*K_SPACE)
#define K_COLOR (-72.13475204444817f)     // -50 * log2(e)
#define K_SPACE (-0.028853900817779268f)  // -(1/50) * log2(e)

__global__ __launch_bounds__(NTHREADS)
void bilateral7x7_kernel(const float* __restrict__ I, float* __restrict__ Out) {
    __shared__ __align__(16) float tile[LH * LSTRIDE];

    const int tid = threadIdx.x;
    const int bx = blockIdx.x, by = blockIdx.y, n = blockIdx.z;

    const float* img = I + (size_t)n * (IMG_H * IMG_W);

    const int gx_al0 = bx * TX - XMARG;  // 16B-aligned column origin (may be negative)
    const int gy0 = by * TY - HALO;

#if defined(__HIP_DEVICE_COMPILE__)
    // Low 32 bits of a flat pointer to LDS are the wave-relative LDS byte
    // offset (addrspacecast(3->0) = {aperture, lds_offset}).
    const uint32_t lds_base = (uint32_t)(uintptr_t)(&tile[0]);
#endif

    // ---- Stage halo tile into LDS as 16B chunks via CDNA5 async DMA ----
    // Image edges (x=0, x=1280) are multiples of 4 floats, and chunk origins
    // are multiples of 4 floats, so each chunk is fully inside or fully
    // outside the image: invalid chunks -> one ds_store_b128 of zeros,
    // valid chunks -> one EXEC-predicated global_load_async_to_lds_b128.
#pragma unroll
    for (int k = 0; k < 2; ++k) {
        const int j = tid + k * NTHREADS;
        if (j < TOTAL_CHUNKS) {
            const int ly = j / NCHUNK;
            const int cx = j - ly * NCHUNK;
            const int gy = gy0 + ly;
            const int gxc = gx_al0 + cx * 4;             // chunk origin column
            const int lidx = ly * LSTRIDE + cx * 4;      // chunk origin in tile
            const bool valid = ((unsigned)gy < (unsigned)IMG_H) &&
                               (gxc >= 0) && (gxc + 4 <= IMG_W);
            if (valid) {
#if defined(__HIP_DEVICE_COMPILE__)
                const uint32_t loff = lds_base + (uint32_t)lidx * 4u;
                const uint32_t goff = (uint32_t)(gy * IMG_W + gxc) * 4u;
                // GVS mode: mem_addr = SADDR(64) + VADDR(32) + inst_offset
                asm volatile("global_load_async_to_lds_b128 %0, %1, %2"
                             :
                             : "v"(loff), "v"(goff), "s"(img)
                             : "memory");
#else
                for (int q = 0; q < 4; ++q)
                    tile[lidx + q] = img[gy * IMG_W + gxc + q];
#endif
            } else {
                float4 z; z.x = 0.f; z.y = 0.f; z.z = 0.f; z.w = 0.f;
                *(float4*)&tile[lidx] = z;               // ds_store_b128
            }
        }
    }
#if defined(__HIP_DEVICE_COMPILE__)
    asm volatile("s_wait_asynccnt 0" ::: "memory");
#endif
    __syncthreads();

    // ---- Compute: each thread produces 4 pixels (64x16 tile / 256 thr) ----
    const int tx = tid & (TX - 1);   // 0..63
    const int ty = tid >> 6;         // 0..3
    const int cxl = tx + XMARG;      // center column inside tile

#pragma unroll
    for (int r = 0; r < TY / 4; ++r) {
        const int oy = ty + 4 * r;   // 0..15
        const float c = tile[(oy + HALO) * LSTRIDE + cxl];

        float wsum = 0.0f;
        float acc  = 0.0f;
#pragma unroll
        for (int dy = 0; dy < 7; ++dy) {
#pragma unroll
            for (int dx = 0; dx < 7; ++dx) {
                const int ry = dy - 3, rx = dx - 3;
                const float cj = (float)(ry * ry + rx * rx) * K_SPACE; // folded const
                const float s = tile[(oy + dy) * LSTRIDE + (cxl - 3 + dx)];
                const float d = s - c;
                const float e = __builtin_fmaf(d * d, K_COLOR, cj);
                const float w = FAST_EXP2(e);   // one v_exp_f32 per tap
                wsum += w;
                acc = __builtin_fmaf(w, s, acc);
            }
        }

        const int gy = by * TY + oy;
        const int gx = bx * TX + tx;
        Out[((size_t)n * IMG_H + gy) * IMG_W + gx] = acc / wsum;
    }
}

extern "C" void kernel_launch(void* const* d_in, const int* in_sizes, int n_in,
                              void* d_out, int out_size, void* d_ws, size_t ws_size,
                              hipStream_t stream) {
    (void)n_in; (void)out_size; (void)d_ws; (void)ws_size;
    const float* I = (const float*)d_in[0];
    float* Out = (float*)d_out;

    const int batch = in_sizes[0] / (IMG_H * IMG_W);  // == 2 for the reference
    dim3 grid(IMG_W / TX, IMG_H / TY, batch > 0 ? batch : 1);
    dim3 block(NTHREADS, 1, 1);
    bilateral7x7_kernel<<<grid, block, 0, stream>>>(I, Out);
}